// CompositeLoss_83691732730136
// MI455X (gfx1250) — compile-verified
//
#include <hip/hip_runtime.h>

typedef __attribute__((ext_vector_type(2))) float v2f;
typedef __attribute__((ext_vector_type(4))) float v4f;
typedef __attribute__((ext_vector_type(8))) float v8f;

// ---------------- workspace layout (floats) ----------------
// [0]=S_dphi [1]=S_dstar [2]=S_phi [3]=S_w1n [4]=S_wn2 [5]=S_det
#define WS_G2 16                 // 32x32 raw gram (h2, n=21 padded)
#define WS_G3 (16 + 32 * 32)     // 80x80 raw gram (h3, n=77 padded)
#define WS_TOTAL (16 + 32 * 32 + 80 * 80)

__global__ void k_zero(float* ws, int n) {
  for (int i = blockIdx.x * blockDim.x + threadIdx.x; i < n;
       i += gridDim.x * blockDim.x)
    ws[i] = 0.0f;
}

__device__ __forceinline__ void block_reduce_atomic(float s, float* acc) {
  __shared__ float red[256];
  red[threadIdx.x] = s;
  __syncthreads();
  for (int off = 128; off > 0; off >>= 1) {
    if (threadIdx.x < off) red[threadIdx.x] += red[threadIdx.x + off];
    __syncthreads();
  }
  if (threadIdx.x == 0) unsafeAtomicAdd(acc, red[0]);
}

__global__ void __launch_bounds__(256) k_sumsq(const float* __restrict__ x,
                                               long long n, float* acc) {
  float s = 0.0f;
  for (long long i = blockIdx.x * (long long)blockDim.x + threadIdx.x; i < n;
       i += (long long)gridDim.x * blockDim.x) {
    float v = x[i];
    s += v * v;
  }
  block_reduce_atomic(s, acc);
}

__global__ void __launch_bounds__(256) k_dot(const float* __restrict__ a,
                                             const float* __restrict__ b,
                                             int n, float* acc) {
  float s = 0.0f;
  for (int i = blockIdx.x * blockDim.x + threadIdx.x; i < n;
       i += gridDim.x * blockDim.x)
    s += a[i] * b[i];
  block_reduce_atomic(s, acc);
}

// 512 independent 7x7 determinants (LU w/ partial pivoting in registers)
__global__ void __launch_bounds__(256) k_det7(const float* __restrict__ m,
                                              int B, float* acc) {
  int b = blockIdx.x * blockDim.x + threadIdx.x;
  float s = 0.0f;
  if (b < B) {
    float a[49];
    for (int i = 0; i < 49; i++) a[i] = m[b * 49 + i];
    float det = 1.0f;
    for (int k = 0; k < 7; k++) {
      int p = k;
      float mx = fabsf(a[k * 7 + k]);
      for (int i = k + 1; i < 7; i++) {
        float v = fabsf(a[i * 7 + k]);
        if (v > mx) { mx = v; p = i; }
      }
      if (p != k) {
        for (int j = 0; j < 7; j++) {
          float t = a[k * 7 + j]; a[k * 7 + j] = a[p * 7 + j]; a[p * 7 + j] = t;
        }
        det = -det;
      }
      float piv = a[k * 7 + k];
      det *= piv;
      float inv = 1.0f / piv;
      for (int i = k + 1; i < 7; i++) {
        float f = a[i * 7 + k] * inv;
        for (int j = k + 1; j < 7; j++) a[i * 7 + j] -= f * a[k * 7 + j];
      }
    }
    float d = det - 1.0f;
    s = d * d;
  }
  block_reduce_atomic(s, acc);
}

// ---------------- WMMA Gram: raw[i,j] = sum_k X[k,i] X[k,j] ----------------
// x: [512, NF, 4096] fp32.  K = 512*4096 reduction dim, N padded to NP.
// LDS tile is [NP][KSTR] (j-major): staging = b128 stores, fragments = b64
// loads.  KSTR=132: 132*16 == 0 (mod 64), 132 == 4 (mod 64) ->
// fragment-load banks = 4*m + 2*khalf + k cover all 64 banks, conflict-free.
template <int NF, int NP, int NT>
__global__ void __launch_bounds__(256) k_gram(const float* __restrict__ x,
                                              float* __restrict__ gacc) {
  constexpr int KT = 128;        // K-slice per iteration (divides 4096)
  constexpr int KSTR = 132;      // LDS row stride (floats)
  constexpr int NPAIRS = NT * (NT + 1) / 2;
  constexpr int NLOAD4 = NP / 8; // float4 staging loads per thread
  __shared__ float tile[NP * KSTR];

  const int tid = threadIdx.x;
  const int lane = tid & 31;
  // wave id as a scalar -> pair selection becomes SALU branches (no exec churn)
  const int wv = __builtin_amdgcn_readfirstlane(tid) >> 5;

  // wave w owns pairs {2w, 2w+1} of the row-major upper-triangle pair list;
  // consecutive pairs often share ti -> A-fragment reuse.
  int np = 0, ti0 = 0, tj0 = 0, ti1 = 0, tj1 = 0;
  {
    const int p0 = 2 * wv, p1 = 2 * wv + 1;
    if (p0 < NPAIRS) {
      np = 1;
      int p = p0, i = 0, row = NT;
      while (p >= row) { p -= row; row--; i++; }
      ti0 = i; tj0 = i + p;
    }
    if (p1 < NPAIRS) {
      np = 2;
      int p = p1, i = 0, row = NT;
      while (p >= row) { p -= row; row--; i++; }
      ti1 = i; tj1 = i + p;
    }
  }
  const bool shareA = (np == 2) && (ti0 == ti1);

  // per-lane fragment base: lane L<16 -> row m=L, K={0,1}; L>=16 -> K={2,3}
  const float* tb = &tile[(lane & 15) * KSTR + 2 * (lane >> 4)];
  const float* pA0 = tb + ti0 * 16 * KSTR;
  const float* pB0 = tb + tj0 * 16 * KSTR;
  const float* pA1 = tb + ti1 * 16 * KSTR;
  const float* pB1 = tb + tj1 * 16 * KSTR;

  // staging: thread t covers rows j = 8*i + (t>>5), k-quad = (t&31)*4
  const int j0 = tid >> 5;           // == wave id: j<NF test is wave-uniform
  const int k4 = (tid & 31) * 4;
  v4f stage[NLOAD4];

  const long long nslices = (512LL * 4096) / KT;
  long long sl = blockIdx.x;

  auto load_slice = [&](long long s) {
    long long kbase = s * KT;
    const float* gbase =
        x + ((long long)(int)(kbase >> 12) * NF + j0) * 4096 +
        (int)(kbase & 4095) + k4;
#pragma unroll
    for (int i = 0; i < NLOAD4; i++) {
      v4f v = {0.0f, 0.0f, 0.0f, 0.0f};
      if (8 * i + j0 < NF) v = *(const v4f*)(gbase + (long long)8 * i * 4096);
      stage[i] = v;
    }
  };

  v8f acc0 = {}; v8f acc1 = {};

  if (sl < nslices) load_slice(sl);
  while (sl < nslices) {
    __syncthreads();  // prior compute finished reading tile
#pragma unroll
    for (int i = 0; i < NLOAD4; i++)
      *(v4f*)&tile[(8 * i + j0) * KSTR + k4] = stage[i];
    __syncthreads();
    const long long nxt = sl + gridDim.x;
    if (nxt < nslices) load_slice(nxt);  // overlaps with WMMA section below

    if (np == 2) {
      if (shareA) {
#pragma unroll 8
        for (int k = 0; k < KT; k += 4) {
          v2f a  = *(const v2f*)(pA0 + k);
          v2f b0 = *(const v2f*)(pB0 + k);
          v2f b1 = *(const v2f*)(pB1 + k);
          acc0 = __builtin_amdgcn_wmma_f32_16x16x4_f32(
              false, a, false, b0, (short)0, acc0, false, false);
          acc1 = __builtin_amdgcn_wmma_f32_16x16x4_f32(
              false, a, false, b1, (short)0, acc1, false, false);
        }
      } else {
#pragma unroll 8
        for (int k = 0; k < KT; k += 4) {
          v2f a0 = *(const v2f*)(pA0 + k);
          v2f b0 = *(const v2f*)(pB0 + k);
          v2f a1 = *(const v2f*)(pA1 + k);
          v2f b1 = *(const v2f*)(pB1 + k);
          acc0 = __builtin_amdgcn_wmma_f32_16x16x4_f32(
              false, a0, false, b0, (short)0, acc0, false, false);
          acc1 = __builtin_amdgcn_wmma_f32_16x16x4_f32(
              false, a1, false, b1, (short)0, acc1, false, false);
        }
      }
    } else if (np == 1) {
#pragma unroll 8
      for (int k = 0; k < KT; k += 4) {
        v2f a0 = *(const v2f*)(pA0 + k);
        v2f b0 = *(const v2f*)(pB0 + k);
        acc0 = __builtin_amdgcn_wmma_f32_16x16x4_f32(
            false, a0, false, b0, (short)0, acc0, false, false);
      }
    }
    sl = nxt;
  }

  // D layout: VGPR r -> M = r (lanes 0-15) / 8+r (lanes 16-31), N = lane&15
  const int nb = lane & 15;
  const int mb = (lane >> 4) * 8;
  if (np >= 1)
    for (int r = 0; r < 8; r++)
      unsafeAtomicAdd(&gacc[(ti0 * 16 + mb + r) * NP + tj0 * 16 + nb], acc0[r]);
  if (np >= 2)
    for (int r = 0; r < 8; r++)
      unsafeAtomicAdd(&gacc[(ti1 * 16 + mb + r) * NP + tj1 * 16 + nb], acc1[r]);
}

// ---------------- final combine (single workgroup) ----------------
__global__ void __launch_bounds__(256) k_final(const float* __restrict__ ws,
                                               float* __restrict__ out) {
  __shared__ float G[77 * 77];
  __shared__ float red[256];
  __shared__ float result[2];
  const int tid = threadIdx.x;

  for (int which = 0; which < 2; which++) {
    const int n = which ? 77 : 21;
    const int NPp = which ? 80 : 32;
    const float* raw = ws + (which ? WS_G3 : WS_G2);
    // symmetrize (only upper tiles were computed) and scale by 1/B
    for (int idx = tid; idx < n * n; idx += 256) {
      int i = idx / n, j = idx % n;
      float v = ((i >> 4) <= (j >> 4)) ? raw[i * NPp + j] : raw[j * NPp + i];
      G[idx] = v * (1.0f / 512.0f);
    }
    __syncthreads();
    // loss_orth = mean((G - I)^2)
    float s = 0.0f;
    for (int idx = tid; idx < n * n; idx += 256) {
      int i = idx / n, j = idx % n;
      float d = G[idx] - (i == j ? 1.0f : 0.0f);
      s += d * d;
    }
    red[tid] = s;
    __syncthreads();
    for (int off = 128; off > 0; off >>= 1) {
      if (tid < off) red[tid] += red[tid + off];
      __syncthreads();
    }
    float loss_orth = red[0] / (float)(n * n);
    __syncthreads();
    // det(G + 1e-6 I) via Gaussian elimination (G ~= I, no pivoting needed)
    for (int i = tid; i < n; i += 256) G[i * n + i] += 1e-6f;
    __syncthreads();
    for (int k = 0; k < n; k++) {
      float inv = 1.0f / G[k * n + k];
      for (int i = k + 1 + tid; i < n; i += 256) {
        float f = G[i * n + k] * inv;
        for (int j = k; j < n; j++) G[i * n + j] -= f * G[k * n + j];
      }
      __syncthreads();
    }
    if (tid == 0) {
      float det = 1.0f;
      for (int k = 0; k < n; k++) det *= G[k * n + k];
      float dd = det - 1.0f;
      result[which] = loss_orth + 0.1f * dd * dd;
    }
    __syncthreads();
  }
  if (tid == 0) {
    float closure = ws[0] / (512.0f * 2401.0f);  // dphi: 7^4 per batch
    float coclos  = ws[1] / (512.0f * 49.0f);    // dstar_phi: 7^2
    float phi2    = ws[2] / (512.0f * 343.0f);   // phi: 7^3
    float w1n     = ws[3] / 512.0f;
    float wn2     = ws[4] / 512.0f;
    float vol     = ws[5] / 512.0f;
    out[0] = closure + coclos + 0.1f * vol + result[0] + result[1] +
             (w1n + wn2) * phi2;
  }
}

extern "C" void kernel_launch(void* const* d_in, const int* in_sizes, int n_in,
                              void* d_out, int out_size, void* d_ws,
                              size_t ws_size, hipStream_t stream) {
  (void)in_sizes; (void)n_in; (void)out_size; (void)ws_size;
  const float* phi    = (const float*)d_in[0];
  const float* dphi   = (const float*)d_in[1];
  const float* dstar  = (const float*)d_in[2];
  const float* metric = (const float*)d_in[4];
  const float* h2     = (const float*)d_in[5];
  const float* h3     = (const float*)d_in[6];
  const float* wm1    = (const float*)d_in[7];
  const float* wneck  = (const float*)d_in[8];
  const float* wm2    = (const float*)d_in[9];
  float* ws = (float*)d_ws;

  k_zero<<<32, 256, 0, stream>>>(ws, WS_TOTAL);
  k_sumsq<<<512, 256, 0, stream>>>(dphi,  512LL * 2401, ws + 0);
  k_sumsq<<<64,  256, 0, stream>>>(dstar, 512LL * 49,   ws + 1);
  k_sumsq<<<128, 256, 0, stream>>>(phi,   512LL * 343,  ws + 2);
  k_dot<<<1, 256, 0, stream>>>(wm1,   wneck, 512, ws + 3);
  k_dot<<<1, 256, 0, stream>>>(wneck, wm2,   512, ws + 4);
  k_det7<<<2, 256, 0, stream>>>(metric, 512, ws + 5);
  // h2: n=21 -> NP=32, NT=2 (3 pairs);  h3: n=77 -> NP=80, NT=5 (15 pairs)
  k_gram<21, 32, 2><<<256, 256, 0, stream>>>(h2, ws + WS_G2);
  k_gram<77, 80, 5><<<1024, 256, 0, stream>>>(h3, ws + WS_G3);
  k_final<<<1, 256, 0, stream>>>(ws, (float*)d_out);
}